// OrthogonalLayer2D_66133906424511
// MI455X (gfx1250) — compile-verified
//
#include <hip/hip_runtime.h>
#include <cstdint>
#include <cstddef>

// Problem constants from the reference: x is [4, 128, 64, 32, 32] f32.
#define M_MODELS 4
#define ROWS     8192          // B*C = 128*64, independent Gram-Schmidt problems
#define D_LEN    1024          // H*W, vector length
#define WPB      4             // waves per block
#define THREADS  (WPB * 32)

typedef uint32_t v4u __attribute__((ext_vector_type(4)));
typedef uint32_t v8u __attribute__((ext_vector_type(8)));
typedef float    f4  __attribute__((ext_vector_type(4)));

__device__ __forceinline__ float wave_sum(float v) {
#pragma unroll
  for (int off = 16; off > 0; off >>= 1)
    v += __shfl_xor(v, off, 32);
  return v;
}

__global__ __launch_bounds__(THREADS)
void OrthogonalLayer2D_gs_kernel(const float* __restrict__ x,
                                 float* __restrict__ out) {
  // 4 waves/block * (4 vectors * 1024 f32) = 64 KB LDS
  __shared__ __align__(16) float lds[WPB][M_MODELS][D_LEN];

  const uint32_t lane = threadIdx.x & 31u;
  const uint32_t wave = (uint32_t)__builtin_amdgcn_readfirstlane(threadIdx.x >> 5);
  const uint32_t row  = blockIdx.x * WPB + wave;   // grid exactly covers ROWS

  // ---------------- TDM async load: 2-D tile (1024 x 4), row stride = M-stride
  // global element offset of x[0, row, :] is row*D_LEN; models are
  // B*C*D = 8,388,608 elements apart.
  const uint64_t ga = (uint64_t)(uintptr_t)x + (uint64_t)row * (D_LEN * 4ull);
  const uint32_t ga_lo = (uint32_t)__builtin_amdgcn_readfirstlane((uint32_t)ga);
  const uint32_t ga_hi = (uint32_t)__builtin_amdgcn_readfirstlane((uint32_t)(ga >> 32));
  const uint32_t lds_off = (uint32_t)__builtin_amdgcn_readfirstlane(
      (uint32_t)(uintptr_t)&lds[wave][0][0]);      // generic addr low 32b = LDS offset

  // D# group 0 (128b): count=1 | lds_addr | global_addr[56:0] | type=2
  v4u g0 = { 1u,
             lds_off,
             ga_lo,
             (ga_hi & 0x01FFFFFFu) | (2u << 30) };

  // D# group 1 (256b):
  //  [17:16] data_size=2 (4B); tensor_dim0=1024; tensor_dim1=4;
  //  tile_dim0=1024; tile_dim1=4; tile_dim2=0;
  //  tensor_dim0_stride = 8,388,608 elems (0x800000); dim1_stride unused.
  v8u g1 = { 0x00020000u,                         // wg_mask=0, data_size=4B
             ((uint32_t)D_LEN & 0xFFFFu) << 16,   // tensor_dim0[15:0]
             ((uint32_t)M_MODELS) << 16,          // tensor_dim1[15:0]
             ((uint32_t)D_LEN) << 16,             // tile_dim0
             (uint32_t)M_MODELS,                  // tile_dim1 (tile_dim2=0)
             0x00800000u,                         // tensor_dim0_stride[31:0]
             0u, 0u };

  asm volatile("tensor_load_to_lds %0, %1" :: "s"(g0), "s"(g1) : "memory");
  asm volatile("s_wait_tensorcnt 0x0" ::: "memory");

  // ---------------- pull row into registers: 4 vectors x 8 float4 per lane
  const f4* lp = (const f4*)&lds[wave][0][0];     // [m*256 + ch*32 + lane]
  f4 a[M_MODELS][8];
#pragma unroll
  for (int m = 0; m < M_MODELS; ++m)
#pragma unroll
    for (int ch = 0; ch < 8; ++ch)
      a[m][ch] = lp[m * 256 + ch * 32 + lane];

  auto dot = [&](const f4* u, const f4* v) -> float {
    float s = 0.f;
#pragma unroll
    for (int ch = 0; ch < 8; ++ch) {
      f4 p = u[ch] * v[ch];
      s += (p.x + p.y) + (p.z + p.w);
    }
    return wave_sum(s);
  };

  // ---------------- classical Gram-Schmidt over the model dimension
  {
    float n0 = sqrtf(dot(a[0], a[0]));
    float i0 = (n0 > 0.f) ? (1.f / n0) : 0.f;     // divide_no_nan
#pragma unroll
    for (int ch = 0; ch < 8; ++ch) a[0][ch] *= i0;
  }

#pragma unroll
  for (int i = 1; i < M_MODELS; ++i) {
    float c[M_MODELS];
#pragma unroll
    for (int j = 0; j < M_MODELS; ++j) {
      if (j < i) c[j] = dot(a[i], a[j]);          // coefs vs. finished basis
      else       c[j] = 0.f;
    }
#pragma unroll
    for (int ch = 0; ch < 8; ++ch) {
      f4 w = a[i][ch];
#pragma unroll
      for (int j = 0; j < M_MODELS; ++j)
        if (j < i) w -= c[j] * a[j][ch];
      a[i][ch] = w;
    }
    float nw = sqrtf(dot(a[i], a[i]));
    float iw = (nw > 0.f) ? (1.f / nw) : 0.f;     // divide_no_nan
#pragma unroll
    for (int ch = 0; ch < 8; ++ch) a[i][ch] *= iw;
  }

  // ---------------- streaming non-temporal stores, perfectly coalesced
#pragma unroll
  for (int m = 0; m < M_MODELS; ++m) {
    f4* op = (f4*)(out + ((size_t)m * ROWS + row) * D_LEN);
#pragma unroll
    for (int ch = 0; ch < 8; ++ch)
      __builtin_nontemporal_store(a[m][ch], op + ch * 32 + lane);
  }
}

extern "C" void kernel_launch(void* const* d_in, const int* in_sizes, int n_in,
                              void* d_out, int out_size, void* d_ws, size_t ws_size,
                              hipStream_t stream) {
  (void)in_sizes; (void)n_in; (void)out_size; (void)d_ws; (void)ws_size;
  const float* x = (const float*)d_in[0];
  float* out     = (float*)d_out;
  dim3 grid(ROWS / WPB);   // 2048 blocks * 4 waves = 8192 rows
  dim3 block(THREADS);
  OrthogonalLayer2D_gs_kernel<<<grid, block, 0, stream>>>(x, out);
}